// IKConstraintLayer_1108101562515
// MI455X (gfx1250) — compile-verified
//
#include <hip/hip_runtime.h>

// IK constraint solver for gfx1250 (MI455X).
// One thread = one kinematic chain. B=64, T=512, N=8, C=3, 50 damped
// pseudo-inverse iterations. Analytic Jacobian (closed form of the reference's
// 1e-6 finite difference), closed-form 2x2 normal-equation pinv (the z-row of
// J is identically zero and joints 3..7 have zero columns, so pinv(J) e
// reduces to J^T (J J^T)^+ e on the xy block).
//
// CDNA5-specific data movement: per-batch bone lengths (reused by 512 chains)
// are staged into LDS with global_load_async_to_lds_b32 (ASYNCcnt path) and
// per-thread streams are touched with global_prefetch.

#define N_SAMPLES (64 * 512)
#define TPB 256

__device__ __forceinline__ float clampf(float x, float lo, float hi) {
    return fminf(fmaxf(x, lo), hi);
}

__global__ __launch_bounds__(TPB)
void ik_constraint_kernel(const float* __restrict__ ang,   // [S, 8] degrees
                          const float* __restrict__ tgt,   // [S, 3]
                          const float* __restrict__ len,   // [64, 3]
                          float* __restrict__ out)         // [S, 8]
{
    __shared__ float sL[4];

    const int tid   = threadIdx.x;
    const int s     = blockIdx.x * TPB + tid;
    const int batch = blockIdx.x >> 1;          // 512 samples/batch, 256/block

    // ---- CDNA5 async-to-LDS staging of the (block-uniform) bone lengths ----
    if (tid < 3) {
        unsigned lds_off = (unsigned)(unsigned long long)(&sL[tid]);
        unsigned long long gaddr =
            (unsigned long long)(len + (size_t)batch * 3 + tid);
        asm volatile("global_load_async_to_lds_b32 %0, %1, off"
                     :: "v"(lds_off), "v"(gaddr)
                     : "memory");
        asm volatile("s_wait_asynccnt 0" ::: "memory");
    }
    __syncthreads();

    const float l0 = sL[0], l1 = sL[1], l2 = sL[2];

    // ---- per-thread input streams (32B-aligned -> b128 vector loads) ----
    __builtin_prefetch(ang + (size_t)s * 8, 0, 0);   // global_prefetch_b8
    __builtin_prefetch(tgt + (size_t)s * 3, 0, 0);

    const float4 av0 = *(const float4*)(ang + (size_t)s * 8);
    const float4 av1 = *(const float4*)(ang + (size_t)s * 8 + 4);
    float a0 = av0.x, a1 = av0.y, a2 = av0.z, a3 = av0.w;
    float a4 = av1.x, a5 = av1.y, a6 = av1.z, a7 = av1.w;

    const float tx = tgt[(size_t)s * 3 + 0];
    const float ty = tgt[(size_t)s * 3 + 1];
    const float tz = tgt[(size_t)s * 3 + 2];

    const float K    = 0.017453292519943295f;   //  pi/180
    const float NK   = -57.295779513082323f;    // -(180/pi)
    const float TOL2 = 1e-4f;                   // 0.01^2
    const float ez2  = tz * tz;                 // z error is constant (pos.z == 0)

    #pragma unroll 1
    for (int it = 0; it < 50; ++it) {
        // FK: cumulative z-rotations of the 3-bone chain
        const float c0 = K * a0;
        const float c1 = c0 + K * a1;
        const float c2 = c1 + K * a2;
        float s0, s1, s2, q0, q1, q2;
        __sincosf(c0, &s0, &q0);
        __sincosf(c1, &s1, &q1);
        __sincosf(c2, &s2, &q2);

        const float u0 = l0 * q0, u1 = l1 * q1, u2 = l2 * q2;   // l*cos
        const float w0 = l0 * s0, w1 = l1 * s1, w2 = l2 * s2;   // l*sin
        const float px = -(w0 + w1 + w2);
        const float py =  (u0 + u1 + u2);
        const float ex = tx - px;
        const float ey = ty - py;

        if (ex * ex + ey * ey + ez2 < TOL2) break;   // == reference freeze

        // Analytic Jacobian columns (suffix sums), common factor -K pulled out:
        //   Jx_j = -K * Sx_j,  Jy_j = -K * Sy_j
        const float Sx2 = u2, Sx1 = u2 + u1, Sx0 = u2 + u1 + u0;
        const float Sy2 = w2, Sy1 = w2 + w1, Sy0 = w2 + w1 + w0;

        // Normal equations on the nonzero 2x2 block of J J^T (H = S S^T):
        const float h00 = Sx0 * Sx0 + Sx1 * Sx1 + Sx2 * Sx2;
        const float h01 = Sx0 * Sy0 + Sx1 * Sy1 + Sx2 * Sy2;
        const float h11 = Sy0 * Sy0 + Sy1 * Sy1 + Sy2 * Sy2;
        const float det = h00 * h11 - h01 * h01;
        const float tr  = h00 + h11;

        float p0, p1;
        if (det > 1e-10f * tr * tr) {
            const float inv = 1.0f / det;                 // full-rank pinv
            p0 = (h11 * ex - h01 * ey) * inv;
            p1 = (h00 * ey - h01 * ex) * inv;
        } else {
            // rank-1 pseudo-inverse: H^+ = H / tr^2 (collinear bones)
            const float inv = (tr > 0.f) ? 1.0f / (tr * tr) : 0.f;
            p0 = (h00 * ex + h01 * ey) * inv;
            p1 = (h01 * ex + h11 * ey) * inv;
        }

        // delta_j = Jx_j*w0 + Jy_j*w1  with w = H^{-1}e / K^2  ->  -(S.p)/K
        a0 = clampf(a0 + NK * (Sx0 * p0 + Sy0 * p1), -90.f,  90.f);
        a1 = clampf(a1 + NK * (Sx1 * p0 + Sy1 * p1),   0.f, 160.f);
        a2 = clampf(a2 + NK * (Sx2 * p0 + Sy2 * p1), -45.f,  45.f);
        // zero-column joints: delta == 0, limits still applied
        a3 = clampf(a3, -45.f,  45.f);
        a4 = clampf(a4,   0.f, 160.f);
        a5 = clampf(a5, -30.f,  30.f);
        a6 = clampf(a6, -30.f,  30.f);
        a7 = clampf(a7, -20.f,  20.f);
    }

    float4 o0; o0.x = a0; o0.y = a1; o0.z = a2; o0.w = a3;
    float4 o1; o1.x = a4; o1.y = a5; o1.z = a6; o1.w = a7;
    *(float4*)(out + (size_t)s * 8)     = o0;
    *(float4*)(out + (size_t)s * 8 + 4) = o1;
}

extern "C" void kernel_launch(void* const* d_in, const int* in_sizes, int n_in,
                              void* d_out, int out_size, void* d_ws, size_t ws_size,
                              hipStream_t stream) {
    const float* ang = (const float*)d_in[0];   // predicted_angles [64,512,8]
    const float* tgt = (const float*)d_in[1];   // target_positions [64,512,3]
    const float* len = (const float*)d_in[2];   // bone_lengths     [64,3]
    float* out = (float*)d_out;                 // [64,512,8]

    dim3 grid(N_SAMPLES / TPB);
    dim3 block(TPB);
    hipLaunchKernelGGL(ik_constraint_kernel, grid, block, 0, stream,
                       ang, tgt, len, out);
}